// Model_28372553957635
// MI455X (gfx1250) — compile-verified
//
#include <hip/hip_runtime.h>

#define N_MAT 100000
#define N_CON 50000
#define NEDGE 1600000
#define NLAB  500000
#define D_MAT 128
#define D_CON 64
#define HID   128
#define OUTD  64

typedef __attribute__((ext_vector_type(2))) float v2f;
typedef __attribute__((ext_vector_type(8))) float v8f;
typedef __attribute__((ext_vector_type(4))) unsigned int u32x4;
typedef __attribute__((ext_vector_type(8))) int i32x8;
typedef __attribute__((ext_vector_type(4))) int i32x4;

#if defined(__has_builtin)
#if __has_builtin(__builtin_amdgcn_tensor_load_to_lds) && \
    __has_builtin(__builtin_amdgcn_s_wait_tensorcnt)
#define HAVE_TDM 1
#endif
#endif

// ---------------------------------------------------------------------------
// TDM: async DMA of a flat fp32 array Global -> LDS (1-D tile descriptor).
// D# group0: count=1 | lds_addr | global_addr[56:0] | type=2
// D# group1: data_size=4B; tensor_dim0 = tile_dim0 = nelem; tensor_dim1 = 1.
// Issue from ONE wave (TDM ignores EXEC); then s_wait_tensorcnt + barrier.
// This toolchain exposes the 6-arg builtin:
//   (u32x4 g0, i32x8 g1, i32x4 g2, i32x4 g3, i32x8 pad, i32 cpol)
// ---------------------------------------------------------------------------
#ifdef HAVE_TDM
__device__ __forceinline__ void tdm_load_f32(const float* gsrc, float* ldst,
                                             unsigned int nelem) {
    unsigned long long ga = (unsigned long long)(uintptr_t)gsrc;
    unsigned int lo = (unsigned int)(uintptr_t)ldst;  // flat low 32 bits == LDS byte offset
    u32x4 g0;
    g0[0] = 1u;                                        // count=1 (valid user descriptor)
    g0[1] = lo;                                        // lds_addr
    g0[2] = (unsigned int)(ga & 0xFFFFFFFFu);          // global_addr[31:0]
    g0[3] = (unsigned int)((ga >> 32) & 0x1FFFFFFu)    // global_addr[56:32]
          | (2u << 30);                                // type = 2 ("image")
    i32x8 g1;
    g1[0] = (int)(2u << 16);                           // data_size = 2 (4 bytes)
    g1[1] = (int)((nelem & 0xFFFFu) << 16);            // tensor_dim0[15:0]
    g1[2] = (int)(((nelem >> 16) & 0xFFFFu)            // tensor_dim0[31:16]
          | (1u << 16));                               // tensor_dim1[15:0] = 1
    g1[3] = (int)((nelem & 0xFFFFu) << 16);            // tensor_dim1 hi=0 | tile_dim0
    g1[4] = 1;                                         // tile_dim1 = 1, tile_dim2 = 0
    g1[5] = (int)nelem;                                // tensor_dim0_stride[31:0]
    g1[6] = (int)((nelem & 0xFFFFu) << 16);            // stride0 hi=0 | stride1[15:0]
    g1[7] = (int)(nelem >> 16);                        // stride1[47:16]
    i32x4 z4 = {0, 0, 0, 0};
    i32x8 z8 = {0, 0, 0, 0, 0, 0, 0, 0};
    __builtin_amdgcn_tensor_load_to_lds(g0, g1, z4, z4, z8, 0);
}
#endif

// ---------------------------------------------------------------------------
// Zero fill (float4 grid-stride)
// ---------------------------------------------------------------------------
__global__ void zero_f4(float4* __restrict__ p, long n4) {
    long i = (long)blockIdx.x * blockDim.x + threadIdx.x;
    long stride = (long)gridDim.x * blockDim.x;
    float4 z = make_float4(0.f, 0.f, 0.f, 0.f);
    for (; i < n4; i += stride) p[i] = z;
}

// ---------------------------------------------------------------------------
// Edge pass 1: one wave per edge.
// ---------------------------------------------------------------------------
__global__ void edge_scatter1(const float* __restrict__ xm, const float* __restrict__ xc,
                              const int* __restrict__ esrc, const int* __restrict__ edst,
                              float* __restrict__ agg_con, float* __restrict__ agg_mat,
                              float* __restrict__ deg_con, float* __restrict__ deg_mat,
                              int E) {
    int wave = (int)((blockIdx.x * blockDim.x + threadIdx.x) >> 5);
    int lane = threadIdx.x & 31;
    if (wave >= E) return;
    int s = esrc[wave];
    int d = edst[wave];
    if (lane == 0) {
        atomicAdd(deg_con + d, 1.0f);
        atomicAdd(deg_mat + s, 1.0f);
    }
    float4 vm = *(const float4*)(xm + (size_t)s * D_MAT + lane * 4);
    float* pd = agg_con + (size_t)d * D_MAT + lane * 4;
    atomicAdd(pd + 0, vm.x); atomicAdd(pd + 1, vm.y);
    atomicAdd(pd + 2, vm.z); atomicAdd(pd + 3, vm.w);
    if (lane < 16) {
        float4 vc = *(const float4*)(xc + (size_t)d * D_CON + lane * 4);
        float* ps = agg_mat + (size_t)s * D_CON + lane * 4;
        atomicAdd(ps + 0, vc.x); atomicAdd(ps + 1, vc.y);
        atomicAdd(ps + 2, vc.z); atomicAdd(ps + 3, vc.w);
    }
}

// ---------------------------------------------------------------------------
// Edge pass 2: both hidden features are 128-wide.
// ---------------------------------------------------------------------------
__global__ void edge_scatter2(const float* __restrict__ hm, const float* __restrict__ hc,
                              const int* __restrict__ esrc, const int* __restrict__ edst,
                              float* __restrict__ agg2_con, float* __restrict__ agg2_mat,
                              int E) {
    int wave = (int)((blockIdx.x * blockDim.x + threadIdx.x) >> 5);
    int lane = threadIdx.x & 31;
    if (wave >= E) return;
    int s = esrc[wave];
    int d = edst[wave];
    float4 vm = *(const float4*)(hm + (size_t)s * HID + lane * 4);
    float* pd = agg2_con + (size_t)d * HID + lane * 4;
    atomicAdd(pd + 0, vm.x); atomicAdd(pd + 1, vm.y);
    atomicAdd(pd + 2, vm.z); atomicAdd(pd + 3, vm.w);
    float4 vc = *(const float4*)(hc + (size_t)d * HID + lane * 4);
    float* ps = agg2_mat + (size_t)s * HID + lane * 4;
    atomicAdd(ps + 0, vc.x); atomicAdd(ps + 1, vc.y);
    atomicAdd(ps + 2, vc.z); atomicAdd(ps + 3, vc.w);
}

// ---------------------------------------------------------------------------
// Fused SAGE combine, register-blocked fp32 WMMA:
//   out[m,:Ho] = act( (agg[m,:]/max(deg,1)) @ Wl^T + b + xs[m,:] @ Wr^T )
// Wave owns 16 rows x HO cols (HO/16 accumulator tiles, K outer loop ->
// each A fragment feeds HO/16 independent WMMAs). Weights staged via TDM.
// n multiple of 16 -> no partial tiles; EXEC full for every WMMA.
// ---------------------------------------------------------------------------
template <int KL, int KR, int HO, bool RELU>
__global__ void sage_combine(const float* __restrict__ agg, const float* __restrict__ deg,
                             const float* __restrict__ Wl, const float* __restrict__ bvec,
                             const float* __restrict__ xs, const float* __restrict__ Wr,
                             float* __restrict__ out, int n) {
    extern __shared__ float smem[];
    float* sWl = smem;            // HO*KL
    float* sWr = smem + HO * KL;  // HO*KR
    int tid = threadIdx.x;
    int wave = tid >> 5, lane = tid & 31;
#ifdef HAVE_TDM
    if (wave == 0) {
        tdm_load_f32(Wl, sWl, HO * KL);
        tdm_load_f32(Wr, sWr, HO * KR);
        __builtin_amdgcn_s_wait_tensorcnt(0);
    }
#else
    for (int i = tid; i < HO * (KL + KR); i += blockDim.x)
        smem[i] = (i < HO * KL) ? Wl[i] : Wr[i - HO * KL];
#endif
    __syncthreads();

    int row0 = (blockIdx.x * 4 + wave) * 16;
    if (row0 >= n) return;  // wave-uniform

    int ln = lane & 15;
    int koff = (lane >> 4) << 1;  // K-pair select per lane half (A & B layouts)
    int m = row0 + ln;
    float scale = 1.0f / fmaxf(deg[m], 1.0f);
    const float* aggRow = agg + (size_t)m * KL;
    const float* xsRow = xs + (size_t)m * KR;

    constexpr int NT = HO / 16;
    v8f acc[NT];
#pragma unroll
    for (int t = 0; t < NT; ++t) {
        float b = bvec[t * 16 + ln];
        acc[t] = (v8f){b, b, b, b, b, b, b, b};
    }

    for (int kb = 0; kb < KL; kb += 4) {
        v2f a;
        a.x = aggRow[kb + koff] * scale;
        a.y = aggRow[kb + koff + 1] * scale;
#pragma unroll
        for (int t = 0; t < NT; ++t) {
            const float* bw = sWl + (size_t)(t * 16 + ln) * KL + kb + koff;
            v2f b;
            b.x = bw[0];
            b.y = bw[1];
            acc[t] = __builtin_amdgcn_wmma_f32_16x16x4_f32(false, a, false, b,
                                                           (short)0, acc[t], false, false);
        }
    }
    for (int kb = 0; kb < KR; kb += 4) {
        v2f a;
        a.x = xsRow[kb + koff];
        a.y = xsRow[kb + koff + 1];
#pragma unroll
        for (int t = 0; t < NT; ++t) {
            const float* bw = sWr + (size_t)(t * 16 + ln) * KR + kb + koff;
            v2f b;
            b.x = bw[0];
            b.y = bw[1];
            acc[t] = __builtin_amdgcn_wmma_f32_16x16x4_f32(false, a, false, b,
                                                           (short)0, acc[t], false, false);
        }
    }

    int hi8 = (lane >> 4) << 3;  // D layout: lanes 16-31 hold M+8
#pragma unroll
    for (int t = 0; t < NT; ++t) {
#pragma unroll
        for (int v = 0; v < 8; ++v) {
            float val = acc[t][v];
            if (RELU) val = fmaxf(val, 0.0f);
            out[(size_t)(row0 + v + hi8) * HO + t * 16 + ln] = val;
        }
    }
}

// ---------------------------------------------------------------------------
// Fused edge decoder: per wave, 16 labels.
//   k = [z_mat[lsrc], z_con[ldst]] -> kout + LDS; h = relu(k@W1^T+b1) (WMMA);
//   logits = h@W2^T+b2; softmax -> probs.  dec_W1 staged via TDM, overlapped
//   with the gather phase.
// Dynamic LDS: sk 4*2048 | sh 4*1024 | sW1 8192 floats.
// ---------------------------------------------------------------------------
#define DEC_WAVES 4
__global__ void decoder(const float* __restrict__ zmat, const float* __restrict__ zcon,
                        const int* __restrict__ lsrc, const int* __restrict__ ldst,
                        const float* __restrict__ W1, const float* __restrict__ b1,
                        const float* __restrict__ W2, const float* __restrict__ b2,
                        float* __restrict__ probs, float* __restrict__ kout, int L) {
    extern __shared__ float dsm[];
    float* sk = dsm;                          // DEC_WAVES * 16 * 128
    float* sh = sk + DEC_WAVES * 16 * 128;    // DEC_WAVES * 16 * 64
    float* sW1 = sh + DEC_WAVES * 16 * 64;    // 64 * 128
    __shared__ int sidx[DEC_WAVES][32];

    int tid = threadIdx.x, wave = tid >> 5, lane = tid & 31;
    int base = (blockIdx.x * DEC_WAVES + wave) * 16;
    bool active = base < L;  // L % 16 == 0 -> full tiles only

#ifdef HAVE_TDM
    if (wave == 0) tdm_load_f32(W1, sW1, OUTD * 128);
#else
    for (int i = tid; i < OUTD * 128; i += blockDim.x) sW1[i] = W1[i];
#endif
    if (active) {
        if (lane < 16) sidx[wave][lane] = lsrc[base + lane];
        else           sidx[wave][lane] = ldst[base + (lane - 16)];
    }
    __syncthreads();
    float* myskw = sk + wave * 16 * 128;
    if (active) {
        for (int i = lane; i < 16 * 128; i += 32) {
            int row = i >> 7, j = i & 127;
            float v = (j < 64) ? zmat[(size_t)sidx[wave][row] * OUTD + j]
                               : zcon[(size_t)sidx[wave][16 + row] * OUTD + (j - 64)];
            myskw[i] = v;
            kout[(size_t)(base + row) * 128 + j] = v;
        }
    }
#ifdef HAVE_TDM
    if (wave == 0) __builtin_amdgcn_s_wait_tensorcnt(0);
#endif
    __syncthreads();

    int ln = lane & 15, koff = (lane >> 4) << 1;
    const float* myk = myskw + ln * 128;
    constexpr int NT = OUTD / 16;  // 4 accumulator tiles
    v8f acc[NT];
#pragma unroll
    for (int t = 0; t < NT; ++t) {
        float b = b1[t * 16 + ln];
        acc[t] = (v8f){b, b, b, b, b, b, b, b};
    }
    for (int kb = 0; kb < 128; kb += 4) {
        v2f a;
        a.x = myk[kb + koff];
        a.y = myk[kb + koff + 1];
#pragma unroll
        for (int t = 0; t < NT; ++t) {
            const float* bw = sW1 + (size_t)(t * 16 + ln) * 128 + kb + koff;
            v2f b;
            b.x = bw[0];
            b.y = bw[1];
            acc[t] = __builtin_amdgcn_wmma_f32_16x16x4_f32(false, a, false, b,
                                                           (short)0, acc[t], false, false);
        }
    }
    float* myshw = sh + wave * 16 * 64;
    int hi8 = (lane >> 4) << 3;
#pragma unroll
    for (int t = 0; t < NT; ++t)
#pragma unroll
        for (int v = 0; v < 8; ++v)
            myshw[(v + hi8) * OUTD + t * 16 + ln] = fmaxf(acc[t][v], 0.0f);
    __syncthreads();

    if (active && lane < 16) {
        const float* hr = myshw + lane * OUTD;
        float d0 = b2[0], d1 = b2[1];
        for (int j = 0; j < OUTD; ++j) {
            float v = hr[j];
            d0 += v * W2[j];
            d1 += v * W2[OUTD + j];
        }
        float mx = fmaxf(d0, d1);
        float e0 = __expf(d0 - mx), e1 = __expf(d1 - mx);
        float s = e0 + e1;
        probs[(size_t)(base + lane) * 2 + 0] = e0 / s;
        probs[(size_t)(base + lane) * 2 + 1] = e1 / s;
    }
}

// ---------------------------------------------------------------------------
extern "C" void kernel_launch(void* const* d_in, const int* in_sizes, int n_in,
                              void* d_out, int out_size, void* d_ws, size_t ws_size,
                              hipStream_t stream) {
    const float* x_mat = (const float*)d_in[0];
    const float* x_con = (const float*)d_in[1];
    const int* esrc = (const int*)d_in[2];
    const int* edst = (const int*)d_in[3];
    const int* lsrc = (const int*)d_in[4];
    const int* ldst = (const int*)d_in[5];
    const float* W1_mc_l = (const float*)d_in[6];
    const float* b1_mc   = (const float*)d_in[7];
    const float* W1_mc_r = (const float*)d_in[8];
    const float* W1_cm_l = (const float*)d_in[9];
    const float* b1_cm   = (const float*)d_in[10];
    const float* W1_cm_r = (const float*)d_in[11];
    const float* W2_mc_l = (const float*)d_in[12];
    const float* b2_mc   = (const float*)d_in[13];
    const float* W2_mc_r = (const float*)d_in[14];
    const float* W2_cm_l = (const float*)d_in[15];
    const float* b2_cm   = (const float*)d_in[16];
    const float* W2_cm_r = (const float*)d_in[17];
    const float* dec_W1  = (const float*)d_in[18];
    const float* dec_b1  = (const float*)d_in[19];
    const float* dec_W2  = (const float*)d_in[20];
    const float* dec_b2  = (const float*)d_in[21];

    // Output layout: probs [L,2], z_mat [N_MAT,64], z_con [N_CON,64], k [L,128]
    float* out   = (float*)d_out;
    float* probs = out;
    float* z_mat = probs + (size_t)NLAB * 2;
    float* z_con = z_mat + (size_t)N_MAT * OUTD;
    float* kout  = z_con + (size_t)N_CON * OUTD;

    // Workspace layout (zero-init prefix first)
    float* ws       = (float*)d_ws;
    float* deg_con  = ws;
    float* deg_mat  = deg_con + N_CON;
    float* agg1_con = deg_mat + N_MAT;                   // N_CON*128
    float* agg1_mat = agg1_con + (size_t)N_CON * D_MAT;  // N_MAT*64
    float* agg2_con = agg1_mat + (size_t)N_MAT * D_CON;  // N_CON*128
    float* agg2_mat = agg2_con + (size_t)N_CON * HID;    // N_MAT*128
    float* h_con    = agg2_mat + (size_t)N_MAT * HID;    // N_CON*128
    float* h_mat    = h_con + (size_t)N_CON * HID;       // N_MAT*128

    long nzero = (long)N_CON + N_MAT + (long)N_CON * D_MAT + (long)N_MAT * D_CON
               + (long)N_CON * HID + (long)N_MAT * HID;
    zero_f4<<<2048, 256, 0, stream>>>((float4*)ws, nzero / 4);

    int eb = (NEDGE + 7) / 8;
    edge_scatter1<<<eb, 256, 0, stream>>>(x_mat, x_con, esrc, edst,
                                          agg1_con, agg1_mat, deg_con, deg_mat, NEDGE);

    // Layer 1 (dynamic LDS up to 96KB; gfx1250 WGP has 320KB)
    int lds1 = HID * (D_MAT + D_CON) * 4;
    (void)hipFuncSetAttribute((const void*)sage_combine<D_MAT, D_CON, HID, true>,
                              hipFuncAttributeMaxDynamicSharedMemorySize, lds1);
    (void)hipFuncSetAttribute((const void*)sage_combine<D_CON, D_MAT, HID, true>,
                              hipFuncAttributeMaxDynamicSharedMemorySize, lds1);
    sage_combine<D_MAT, D_CON, HID, true>
        <<<(N_CON + 63) / 64, 128, lds1, stream>>>(
            agg1_con, deg_con, W1_mc_l, b1_mc, x_con, W1_mc_r, h_con, N_CON);
    sage_combine<D_CON, D_MAT, HID, true>
        <<<(N_MAT + 63) / 64, 128, lds1, stream>>>(
            agg1_mat, deg_mat, W1_cm_l, b1_cm, x_mat, W1_cm_r, h_mat, N_MAT);

    edge_scatter2<<<eb, 256, 0, stream>>>(h_mat, h_con, esrc, edst,
                                          agg2_con, agg2_mat, NEDGE);

    // Layer 2 (writes straight into d_out)
    int lds2 = OUTD * (HID + HID) * 4;
    sage_combine<HID, HID, OUTD, false>
        <<<(N_CON + 63) / 64, 128, lds2, stream>>>(
            agg2_con, deg_con, W2_mc_l, b2_mc, h_con, W2_mc_r, z_con, N_CON);
    sage_combine<HID, HID, OUTD, false>
        <<<(N_MAT + 63) / 64, 128, lds2, stream>>>(
            agg2_mat, deg_mat, W2_cm_l, b2_cm, h_mat, W2_cm_r, z_mat, N_MAT);

    // Decoder (gather + concat + MLP + softmax, fused)
    int ldsd = (DEC_WAVES * 16 * 128 + DEC_WAVES * 16 * 64 + OUTD * 128) * 4;
    (void)hipFuncSetAttribute((const void*)decoder,
                              hipFuncAttributeMaxDynamicSharedMemorySize, ldsd);
    decoder<<<(NLAB + 63) / 64, 128, ldsd, stream>>>(z_mat, z_con, lsrc, ldst,
                                                     dec_W1, dec_b1, dec_W2, dec_b2,
                                                     probs, kout, NLAB);
}